// MCSM_45071386805057
// MI455X (gfx1250) — compile-verified
//
#include <hip/hip_runtime.h>
#include <stdint.h>

// ---------------------------------------------------------------------------
// Monte-Carlo softmax expectation over channel dim.
//   out[b,c,h,w] = (1/32) * sum_i softmax_c( logits + eps_i * std )
// Shape (8, 19, 512, 512), channel stride = 512*512 floats.
//
// CDNA5 path: TENSOR_LOAD_TO_LDS (TDM) gathers the 19-row strided channel
// tile (stride 1MB) into LDS with a single descriptor per tensor, tracked on
// TENSORcnt. Compute is trans-bound (v_exp_f32), fp32 throughout.
// ---------------------------------------------------------------------------

#define NCH   19
#define TILE  256
#define HW    (512 * 512)
#define NSAMP 32

typedef unsigned int v4u __attribute__((ext_vector_type(4)));
typedef int          v8i __attribute__((ext_vector_type(8)));
typedef int          v4i __attribute__((ext_vector_type(4)));

__device__ __forceinline__ uint32_t pcg_hash(uint32_t v) {
    uint32_t s = v * 747796405u + 2891336453u;
    uint32_t w = ((s >> ((s >> 28) + 4u)) ^ s) * 277803737u;
    return (w >> 22) ^ w;
}

// Build a 2D-tile Tensor DMA Descriptor (D#) and issue TENSOR_LOAD_TO_LDS.
//   tile: tile_d0 x tile_d1 elements (4B each), row stride = stride0 elements.
// Field packing per CDNA5 ISA (group0) / §8.4 (group1).
__device__ __forceinline__ void tdm_load_tile_2d(uint32_t lds_byte_off,
                                                 const void* gptr,
                                                 uint32_t tile_d0, uint32_t tile_d1,
                                                 uint32_t tdim0, uint32_t tdim1,
                                                 uint64_t stride0_elems) {
    const uint64_t ga = (uint64_t)(uintptr_t)gptr;

    v4u g0;
    g0.x = 0x1u;                                  // count=1 (valid), user mode
    g0.y = lds_byte_off;                          // lds_addr [63:32]
    g0.z = (uint32_t)(ga & 0xFFFFFFFFull);        // global_addr[31:0] -> bits 95:64
    g0.w = (uint32_t)((ga >> 32) & 0x01FFFFFFull) // global_addr[56:32] -> bits 120:96
         | (2u << 30);                            // type=2 ("image")

    v8i g1;
    g1[0] = (int)(2u << 16);                      // workgroup_mask=0, data_size=2 (4B)
    g1[1] = (int)((tdim0 & 0xFFFFu) << 16);       // tensor_dim0[15:0] @ bits 63:48
    g1[2] = (int)(((tdim0 >> 16) & 0xFFFFu)       // tensor_dim0[31:16]
         |        ((tdim1 & 0xFFFFu) << 16));     // tensor_dim1[15:0]
    g1[3] = (int)(((tdim1 >> 16) & 0xFFFFu)       // tensor_dim1[31:16]
         |        ((tile_d0 & 0xFFFFu) << 16));   // tile_dim0
    g1[4] = (int)(tile_d1 & 0xFFFFu);             // tile_dim1 ; tile_dim2=0 (unused)
    g1[5] = (int)(uint32_t)(stride0_elems & 0xFFFFFFFFull);        // dim0_stride[31:0]
    g1[6] = (int)(uint32_t)((stride0_elems >> 32) & 0xFFFFull);    // dim0_stride[47:32]
    g1[7] = 0;                                                     // dim1_stride hi = 0

    const v4i gz = {0, 0, 0, 0};
#if __has_include(<hip/amd_detail/amd_gfx1250_TDM.h>)
    const v8i gz8 = {0, 0, 0, 0, 0, 0, 0, 0};
    __builtin_amdgcn_tensor_load_to_lds(g0, g1, gz, gz, gz8, 0);   // clang-23 6-arg form
#else
    __builtin_amdgcn_tensor_load_to_lds(g0, g1, gz, gz, 0);        // ROCm 7.2 5-arg form
#endif
}

__global__ __launch_bounds__(TILE)
void MCSM_45071386805057_kernel(const float* __restrict__ logits,
                                const float* __restrict__ stddev,
                                float* __restrict__ out) {
    __shared__ float sm_l[NCH * TILE];
    __shared__ float sm_s[NCH * TILE];

    const int      tid   = threadIdx.x;
    const uint32_t tile  = blockIdx.x;            // 8192 tiles of 256 pixels
    const uint32_t batch = tile / (HW / TILE);    // 1024 tiles per batch image
    const uint32_t hw0   = (tile % (HW / TILE)) * TILE;
    const size_t   gbase = (size_t)batch * (size_t)NCH * HW + hw0;

    // Wave 0 issues both TDM tile loads (EXEC is ignored by TDM; other waves
    // branch around the instruction entirely), drains TENSORcnt, then the
    // workgroup barrier publishes the LDS tiles to all 8 waves.
    if (tid < 32) {
        tdm_load_tile_2d((uint32_t)(uintptr_t)&sm_l[0], logits + gbase,
                         TILE, NCH, HW, NCH, (uint64_t)HW);
        tdm_load_tile_2d((uint32_t)(uintptr_t)&sm_s[0], stddev + gbase,
                         TILE, NCH, HW, NCH, (uint64_t)HW);
        __builtin_amdgcn_s_wait_tensorcnt(0);
    }
    __syncthreads();

    float l[NCH], s[NCH], acc[NCH];
#pragma unroll
    for (int c = 0; c < NCH; ++c) {
        l[c]   = sm_l[c * TILE + tid];
        s[c]   = sm_s[c * TILE + tid];
        acc[c] = 0.0f;
    }

    const uint32_t pix = tile * TILE + (uint32_t)tid;

#pragma unroll 1
    for (int i = 0; i < NSAMP; ++i) {
        const uint32_t seed = pix * (NSAMP * NCH) + (uint32_t)i * NCH;
        float e[NCH];
        float sum = 0.0f;
#pragma unroll
        for (int c = 0; c < NCH; ++c) {
            // CLT Gaussian: 4x uniform16 from two PCG hashes -> N(0,1) approx.
            const uint32_t h1 = pcg_hash(seed + (uint32_t)c);
            const uint32_t h2 = pcg_hash(h1 ^ 0x9E3779B9u);
            const uint32_t S  = (h1 & 0xFFFFu) + (h1 >> 16)
                              + (h2 & 0xFFFFu) + (h2 >> 16);
            const float eps = (float)(int)S * (1.7320508f / 65536.0f) - 3.4641016f;

            const float x  = fmaf(eps, s[c], l[c]);
            const float ec = __builtin_amdgcn_exp2f(x * 1.44269504f); // v_exp_f32
            e[c] = ec;
            sum += ec;
        }
        const float r = __builtin_amdgcn_rcpf(sum);                  // v_rcp_f32
#pragma unroll
        for (int c = 0; c < NCH; ++c)
            acc[c] = fmaf(e[c], r, acc[c]);
    }

    const float inv_n = 1.0f / (float)NSAMP;
#pragma unroll
    for (int c = 0; c < NCH; ++c)
        out[gbase + (size_t)c * HW + tid] = acc[c] * inv_n;
}

extern "C" void kernel_launch(void* const* d_in, const int* in_sizes, int n_in,
                              void* d_out, int out_size, void* d_ws, size_t ws_size,
                              hipStream_t stream) {
    const float* logits = (const float*)d_in[0];
    const float* stddev = (const float*)d_in[1];
    float*       out    = (float*)d_out;

    const int total_pixels = 8 * HW;          // 2,097,152 pixels
    const int blocks       = total_pixels / TILE;  // 8192

    MCSM_45071386805057_kernel<<<blocks, TILE, 0, stream>>>(logits, stddev, out);
}